// RandAugUDA_2173253452143
// MI455X (gfx1250) — compile-verified
//
#include <hip/hip_runtime.h>

// Shapes from the reference: x = (256, 3, 224, 224) fp32
#define CHW        150528      // 3*224*224
#define V4_PER_IMG 37632       // CHW / 4
#define BLK        256
#define BLOCKS_PER_IMG 147     // V4_PER_IMG / BLK

typedef __attribute__((ext_vector_type(2))) float v2f;
typedef __attribute__((ext_vector_type(4))) float v4f;
typedef __attribute__((ext_vector_type(8))) float v8f;

// ---------------------------------------------------------------------------
// Kernel 1: per-image mean. One block (8 wave32) per image.
// Per-thread float4 accumulation, then wave reduction via V_WMMA_F32_16X16X4_F32
// with B = ones (sum-by-matmul), then tiny LDS combine across the 8 waves.
// ---------------------------------------------------------------------------
__global__ void __launch_bounds__(BLK)
mean_kernel(const float* __restrict__ x, float* __restrict__ mean_out) {
  const int b = blockIdx.x;
  const v4f* __restrict__ xi = (const v4f*)(x + (size_t)b * CHW);

  float acc = 0.0f;
  #pragma unroll 4
  for (int i = threadIdx.x; i < V4_PER_IMG; i += BLK) {
    v4f v = xi[i];
    acc += (v.x + v.y) + (v.z + v.w);
  }

  // --- wave32 reduction via WMMA: D = A(16x4) * ones(4x16) + 0 ---
  // A layout (32-bit, 16x4): VGPR0 lanes 0-15 -> (M=lane, K=0),
  // lanes 16-31 -> (M=lane-16, K=2). VGPR1 = K=1/K=3 (set to 0).
  // => D[m][n] = p[m] + p[m+16], replicated across all 16 columns.
  v2f A; A.x = acc;  A.y = 0.0f;
  v2f Bm; Bm.x = 1.0f; Bm.y = 1.0f;
  v8f C = {};
  C = __builtin_amdgcn_wmma_f32_16x16x4_f32(
      /*neg_a=*/false, A, /*neg_b=*/false, Bm,
      /*c_mod=*/(short)0, C, /*reuse_a=*/false, /*reuse_b=*/false);

  // Each lane holds 8 row-values of its column: lanes 0-15 rows 0..7,
  // lanes 16-31 rows 8..15. Sum them, then fold halves with xor-16 shuffle.
  float s = ((C[0] + C[1]) + (C[2] + C[3])) + ((C[4] + C[5]) + (C[6] + C[7]));
  s += __shfl_xor(s, 16, 32);   // now every lane has the full wave sum

  __shared__ float wsum[BLK / 32];
  const int lane = threadIdx.x & 31;
  const int wid  = threadIdx.x >> 5;
  if (lane == 0) wsum[wid] = s;
  __syncthreads();
  if (threadIdx.x == 0) {
    float t = 0.0f;
    #pragma unroll
    for (int w = 0; w < BLK / 32; ++w) t += wsum[w];
    mean_out[b] = t * (1.0f / (float)CHW);
  }
}

// ---------------------------------------------------------------------------
// Kernel 2: elementwise transform. Blocks are image-aligned so sample/mean/
// tf selection are wave-uniform (scalar loads + scalar branches).
// Output stored non-temporally to keep x resident in the 192MB L2.
// ---------------------------------------------------------------------------
__device__ __forceinline__ float clip01(float v) {
  return fminf(fmaxf(v, 0.0f), 1.0f);
}

__global__ void __launch_bounds__(BLK)
aug_kernel(const float* __restrict__ x,
           const int*   __restrict__ sample,
           const int*   __restrict__ apply_mask,
           const float* __restrict__ mean_in,
           float* __restrict__ out) {
  const int blk = blockIdx.x;
  const int b   = blk / BLOCKS_PER_IMG;
  const int i   = (blk - b * BLOCKS_PER_IMG) * BLK + threadIdx.x;
  const size_t base = (size_t)b * CHW + (size_t)i * 4;

  const int   s   = sample[b];                 // uniform per block
  const int   am  = apply_mask[s];             // uniform per block
  const int   tf  = s / 9;                     // uniform per block
  const float mg  = (float)(s - tf * 9 + 1) * 0.1f;
  const float mu  = mean_in[b];

  v4f v = *(const v4f*)(x + base);
  v4f r;

  if (am > 0) {
    if (tf == 0) {            // brightness: clip(x + mag)
      r.x = clip01(v.x + mg); r.y = clip01(v.y + mg);
      r.z = clip01(v.z + mg); r.w = clip01(v.w + mg);
    } else if (tf == 1) {     // contrast: clip(mean + (x-mean)*(1+mag))
      const float g = 1.0f + mg;
      r.x = clip01(fmaf(v.x - mu, g, mu));
      r.y = clip01(fmaf(v.y - mu, g, mu));
      r.z = clip01(fmaf(v.z - mu, g, mu));
      r.w = clip01(fmaf(v.w - mu, g, mu));
    } else if (tf == 2) {     // invert blend: (1-mag)*x + mag*(1-x)
      const float a = 1.0f - mg;
      r.x = a * v.x + mg * (1.0f - v.x);
      r.y = a * v.y + mg * (1.0f - v.y);
      r.z = a * v.z + mg * (1.0f - v.z);
      r.w = a * v.w + mg * (1.0f - v.w);
    } else {                  // gain: clip(x*(1+mag))
      const float g = 1.0f + mg;
      r.x = clip01(v.x * g); r.y = clip01(v.y * g);
      r.z = clip01(v.z * g); r.w = clip01(v.w * g);
    }
  } else {
    r = v;
  }

  __builtin_nontemporal_store(r, (v4f*)(out + base));
}

// ---------------------------------------------------------------------------
extern "C" void kernel_launch(void* const* d_in, const int* in_sizes, int n_in,
                              void* d_out, int out_size, void* d_ws, size_t ws_size,
                              hipStream_t stream) {
  const float* x          = (const float*)d_in[0];
  const int*   sample     = (const int*)d_in[1];
  const int*   apply_mask = (const int*)d_in[2];
  float*       out        = (float*)d_out;
  float*       means      = (float*)d_ws;          // B floats of scratch

  const int B = in_sizes[1];                       // 256 images

  mean_kernel<<<B, BLK, 0, stream>>>(x, means);
  aug_kernel<<<B * BLOCKS_PER_IMG, BLK, 0, stream>>>(x, sample, apply_mask, means, out);
}